// yolo_target_35381940584553
// MI455X (gfx1250) — compile-verified
//
#include <hip/hip_runtime.h>

#define NCH 85
#define CONF_TH 1e-4f

typedef __attribute__((ext_vector_type(2))) float v2f;
typedef __attribute__((ext_vector_type(8))) float v8f;

// ---------------------------------------------------------------------------
// ws layout: [0..3]  : unsigned cutoff (first row with conf < CONF, ~0u if none)
//            [16..]  : float partials[REDUCE_BLOCKS]
// ---------------------------------------------------------------------------

__global__ void init_ws_kernel(unsigned* __restrict__ cut) {
    *cut = 0xFFFFFFFFu;
}

// Pass 1: find first failing row (order-independent atomicMin -> deterministic).
__global__ __launch_bounds__(256) void find_cutoff_kernel(
        const float* __restrict__ data, int k, unsigned* __restrict__ cut) {
    int i = blockIdx.x * blockDim.x + threadIdx.x;
    const int stride = gridDim.x * blockDim.x;
    unsigned local = 0xFFFFFFFFu;
    for (; i < k; i += stride) {
        const float c = data[(size_t)i * NCH + 4];
        if (c < CONF_TH) local = min(local, (unsigned)i);
    }
    if (local != 0xFFFFFFFFu) atomicMin(cut, local);
}

// Pass 2: gated reduction. Box-sum (channels 0..3) done with
// v_wmma_f32_16x16x4_f32 against an all-ones B; conf accumulated in VALU.
// A-matrix layout (ISA 7.12.2, 32-bit A 16x4): lanes 0-15 hold K=0 (v0) and
// K=1 (v1) for rows M=0..15; lanes 16-31 hold K=2,K=3 for the same rows.
__global__ __launch_bounds__(256) void reduce_wmma_kernel(
        const float* __restrict__ data, int k,
        const unsigned* __restrict__ cutp, float* __restrict__ partials) {
    const unsigned cut = min(*cutp, (unsigned)k);

    const int lane          = threadIdx.x & 31;
    const int sub           = lane & 15;              // row within 16-row tile
    const int chOff         = (lane < 16) ? 0 : 2;    // K=0,1 vs K=2,3
    const int waveInBlock   = threadIdx.x >> 5;
    const int wavesPerBlock = blockDim.x >> 5;        // 8 (wave32)
    const int waveId        = blockIdx.x * wavesPerBlock + waveInBlock;
    const int nWaves        = gridDim.x * wavesPerBlock;
    const int rowStep       = nWaves * 16;            // rows per grid iteration

    v8f acc = {0.f, 0.f, 0.f, 0.f, 0.f, 0.f, 0.f, 0.f};
    float confAcc = 0.f;
    const v2f ones = {1.f, 1.f};

    for (int base = waveId * 16; base < k; base += rowStep) {
        const int r = base + sub;
        v2f a = {0.f, 0.f};
        float cf = 0.f;
        if ((unsigned)r < cut) {
            const float* row = data + (size_t)r * NCH;
            a.x = row[chOff];
            a.y = row[chOff + 1];
            if (lane < 16) cf = row[4];
        }
        // Prefetch next chunk's line for this lane (global_prefetch_b8).
        const int rn = r + rowStep;
        if ((unsigned)rn < cut)
            __builtin_prefetch(data + (size_t)rn * NCH + chOff, 0, 0);

        // D[i][j] += sum_k A[i][k]  (replicated across the 16 columns j)
        acc = __builtin_amdgcn_wmma_f32_16x16x4_f32(
                /*neg_a=*/false, a, /*neg_b=*/false, ones,
                /*c_mod=*/(short)0, acc, /*reuse_a=*/false, /*reuse_b=*/false);
        confAcc += cf;
    }

    // Sum of all lanes over all 8 C VGPRs == 16 * (total box sum).
    float s = acc[0] + acc[1] + acc[2] + acc[3] +
              acc[4] + acc[5] + acc[6] + acc[7];
    float w = s * 0.0625f + confAcc;

    // wave32 butterfly reduction
    for (int off = 16; off > 0; off >>= 1)
        w += __shfl_xor(w, off, 32);

    __shared__ float smem[8];
    if (lane == 0) smem[waveInBlock] = w;
    __syncthreads();
    if (threadIdx.x == 0) {
        float b = 0.f;
        for (int i = 0; i < wavesPerBlock; ++i) b += smem[i];
        partials[blockIdx.x] = b;
    }
}

// Deterministic serial finalize over block partials.
__global__ void finalize_kernel(const float* __restrict__ partials, int n,
                                float* __restrict__ out) {
    if (threadIdx.x == 0 && blockIdx.x == 0) {
        float s = 0.f;
        for (int i = 0; i < n; ++i) s += partials[i];
        out[0] = s;
    }
}

extern "C" void kernel_launch(void* const* d_in, const int* in_sizes, int n_in,
                              void* d_out, int out_size, void* d_ws, size_t ws_size,
                              hipStream_t stream) {
    (void)n_in; (void)out_size; (void)ws_size;

    const float* data = (const float*)d_in[0];
    const int n = in_sizes[0] / NCH;   // 1048576
    const int k = n / 2;               // RATIO = 0.5 -> 524288

    unsigned* d_cut     = (unsigned*)d_ws;
    float*   d_partials = (float*)((char*)d_ws + 16);

    const int REDUCE_BLOCKS = 512;     // 4096 waves, 8 grid iterations over k

    init_ws_kernel<<<1, 1, 0, stream>>>(d_cut);
    find_cutoff_kernel<<<1024, 256, 0, stream>>>(data, k, d_cut);
    reduce_wmma_kernel<<<REDUCE_BLOCKS, 256, 0, stream>>>(data, k, d_cut, d_partials);
    finalize_kernel<<<1, 1, 0, stream>>>(d_partials, REDUCE_BLOCKS, (float*)d_out);
}